// DotPredictor_5179730559506
// MI455X (gfx1250) — compile-verified
//
#include <hip/hip_runtime.h>

typedef __attribute__((ext_vector_type(2))) float v2f;
typedef __attribute__((ext_vector_type(8))) float v8f;

// Per-edge dot product via V_WMMA_F32_16X16X4_F32 diagonal trick.
// One wave32 handles a tile of 16 edges:
//   A (16x4 f32) = u rows of the 16 edges for K-slice [4t, 4t+3]
//   B (4x16 f32) = v rows of the SAME 16 edges for the same K-slice
//   D[i][j] += sum_k u_i[k] * v_j[k]   -> diagonal = the 16 edge dots
// A/B per-lane layout (ISA 7.12.2, 32-bit A 16x4): lane L<16 holds
// (M=L, K=4t..4t+1), lane L+16 holds (M=L, K=4t+2..4t+3) -> each lane does one
// float2 load from its edge's u row and one from its v row per K-step.
__global__ __launch_bounds__(256, 1) void edge_dot_wmma_f32(
    const float* __restrict__ emb,
    const long long* __restrict__ src,
    const long long* __restrict__ dst,
    float* __restrict__ out,
    int n_edges)
{
  const int lane        = threadIdx.x & 31;
  const int waveInBlock = threadIdx.x >> 5;
  const int eSub        = lane & 15;      // which of the 16 edges this lane feeds
  const bool hi         = lane >= 16;     // upper half-wave -> K offset +2

  const long long tile = (long long)blockIdx.x * 8 + waveInBlock; // 16 edges/tile
  long long e = tile * 16 + eSub;
  // Clamp (not mask) out-of-range edges: EXEC must stay all-1s for WMMA.
  const long long eClamped = (e < (long long)n_edges) ? e : (long long)(n_edges - 1);

  const long long s = src[eClamped];
  const long long d = dst[eClamped];

  // 128 floats per row -> row base = idx << 7; half-wave K offset 0 or 2.
  const float* uRow = emb + (s << 7) + (hi ? 2 : 0);
  const float* vRow = emb + (d << 7) + (hi ? 2 : 0);

  v8f acc = {};
#pragma unroll
  for (int t = 0; t < 32; ++t) {
    v2f a = *(const v2f*)(uRow + 4 * t);   // u[e][4t + (hi?2:0) .. +1]
    v2f b = *(const v2f*)(vRow + 4 * t);   // v[e][4t + (hi?2:0) .. +1]
    // 8 args: (neg_a, A, neg_b, B, c_mod, C, reuse_a, reuse_b)
    acc = __builtin_amdgcn_wmma_f32_16x16x4_f32(
        false, a, false, b, (short)0, acc, false, false);
  }

  // Diagonal lives at: edge M<8  -> VGPR M,   lane M      (lanes 0..7)
  //                    edge M>=8 -> VGPR M-8, lane M+16   (lanes 24..31)
  const int sel = (hi ? (lane - 24) : lane) & 7;
  float x0 = (sel & 1) ? acc[1] : acc[0];
  float x1 = (sel & 1) ? acc[3] : acc[2];
  float x2 = (sel & 1) ? acc[5] : acc[4];
  float x3 = (sel & 1) ? acc[7] : acc[6];
  float y0 = (sel & 2) ? x1 : x0;
  float y1 = (sel & 2) ? x3 : x2;
  float r  = (sel & 4) ? y1 : y0;

  const bool active  = (lane < 8) || (lane >= 24);
  const long long eo = tile * 16 + (hi ? (lane - 16) : lane);
  if (active && eo < (long long)n_edges) {
    __builtin_nontemporal_store(r, out + eo);  // streaming write, don't pollute caches
  }
}

extern "C" void kernel_launch(void* const* d_in, const int* in_sizes, int n_in,
                              void* d_out, int out_size, void* d_ws, size_t ws_size,
                              hipStream_t stream) {
  const float*     emb = (const float*)d_in[0];
  const long long* src = (const long long*)d_in[1];
  const long long* dst = (const long long*)d_in[2];
  float* out = (float*)d_out;

  const int n_edges = in_sizes[1];          // E = number of edges
  const int tiles   = (n_edges + 15) / 16;  // 16 edges per wave
  const int blocks  = (tiles + 7) / 8;      // 8 waves per 256-thread block

  edge_dot_wmma_f32<<<blocks, 256, 0, stream>>>(emb, src, dst, out, n_edges);
}